// ExampleLabelWeights_58377195487799
// MI455X (gfx1250) — compile-verified
//
#include <hip/hip_runtime.h>
#include <hip/hip_bf16.h>

#define MAX_CARD 16

typedef float v2f __attribute__((ext_vector_type(2)));
typedef float v4f __attribute__((ext_vector_type(4)));
typedef float v8f __attribute__((ext_vector_type(8)));

__global__ void ExampleLabelWeights_zero_kernel(float* out) {
    out[0] = 0.0f;
}

// ds_swizzle_b32 group-of-32 XOR: offset[15]=0, xor_mask=[14:10], and_mask=[4:0]=0x1f
// Immediate must be a compile-time constant -> template parameter.
template <int XOR_MASK>
__device__ __forceinline__ float swizzle_xor_add(float v) {
    int s = __builtin_amdgcn_ds_swizzle(__float_as_int(v), (XOR_MASK << 10) | 0x1f);
    return v + __int_as_float(s);
}

// One thread per batch row. Wave32: 32 rows per wave.
// Per-row: gather params row (64B, L2-resident), stream losses row (64B, NT),
// branchless masked softmax, dot with losses -> per-lane partial p.
// Wave reduction: V_WMMA_F32_16X16X4_F32 (A = ones 16x4, B vgpr0 = p) gives
// colsum[n] = p[n] + p[n+16] in every lane (n = lane%16); a 4-step ds_swizzle
// XOR butterfly over the low 4 lane bits finishes the 16-way column sum.
__global__ __launch_bounds__(256) void ExampleLabelWeights_58377195487799_kernel(
    const float* __restrict__ losses,       // [B, 16]
    const int*   __restrict__ inputs_idx,   // [B]
    const float* __restrict__ params,       // [N, 16]
    const int*   __restrict__ cardinality,  // [N]
    float* __restrict__ out,                // [1]
    int B)
{
    const int b = blockIdx.x * blockDim.x + threadIdx.x;

    float p = 0.0f;
    if (b < B) {
        const int idx  = inputs_idx[b];
        const int card = cardinality[idx];

        // 64-byte aligned 128-bit vector loads (rows are 16 floats = 64 B).
        const v4f* wrow = reinterpret_cast<const v4f*>(params + (size_t)idx * MAX_CARD);
        const v4f* lrow = reinterpret_cast<const v4f*>(losses + (size_t)b   * MAX_CARD);

        float w[MAX_CARD], l[MAX_CARD];
#pragma unroll
        for (int i = 0; i < 4; ++i) {
            v4f wv = wrow[i];                              // regular: keep params hot in L2
            v4f lv = __builtin_nontemporal_load(lrow + i); // streamed once: NT hint
#pragma unroll
            for (int j = 0; j < 4; ++j) {
                w[4*i + j] = wv[j];
                l[4*i + j] = lv[j];
            }
        }

        // Branchless masked softmax over slots [0, card), then weighted loss sum.
        float m = -__builtin_inff();
#pragma unroll
        for (int k = 0; k < MAX_CARD; ++k)
            m = fmaxf(m, (k < card) ? w[k] : -__builtin_inff());

        float s = 0.0f, acc = 0.0f;
#pragma unroll
        for (int k = 0; k < MAX_CARD; ++k) {
            float e = __expf(w[k] - m);        // always execute (TRANS op), then mask
            e = (k < card) ? e : 0.0f;         // v_cndmask, no divergence
            s   += e;
            acc  = fmaf(e, l[k], acc);
        }
        p = acc * __builtin_amdgcn_rcpf(s);    // v_rcp_f32; card >= 1 so s >= 1
    }

    // --- wave reduction on the matrix pipe (uniform control flow; EXEC all 1s) ---
    // A = ones(16x4) (both A VGPRs = 1.0); B vgpr0 = p, vgpr1 = 0: each lane's
    // value occupies a unique (k, n=lane%16) slot of the 4x16 B tile, so
    // D[i][n] = p[n] + p[n+16] independent of the exact K-row assignment.
    v2f av; av[0] = 1.0f; av[1] = 1.0f;
    v2f bv; bv[0] = p;    bv[1] = 0.0f;
    v8f cacc = {};
    cacc = __builtin_amdgcn_wmma_f32_16x16x4_f32(
        /*neg_a=*/false, av, /*neg_b=*/false, bv,
        /*c_mod=*/(short)0, cacc, /*reuse_a=*/false, /*reuse_b=*/false);

    // Every lane holds colsum[lane % 16] in cacc[0] (all D rows identical).
    // XOR butterfly over low 4 lane bits -> lane 0 = sum of 16 distinct columns
    // = sum of all 32 lane partials.
    float v = cacc[0];
    v = swizzle_xor_add<1>(v);
    v = swizzle_xor_add<2>(v);
    v = swizzle_xor_add<4>(v);
    v = swizzle_xor_add<8>(v);

    if ((threadIdx.x & 31) == 0)
        atomicAdd(out, v);
}

extern "C" void kernel_launch(void* const* d_in, const int* in_sizes, int n_in,
                              void* d_out, int out_size, void* d_ws, size_t ws_size,
                              hipStream_t stream) {
    const float* losses      = (const float*)d_in[0];
    const int*   inputs_idx  = (const int*)  d_in[1];
    const float* params      = (const float*)d_in[2];
    const int*   cardinality = (const int*)  d_in[3];
    float* out = (float*)d_out;

    const int B = in_sizes[1];              // BATCH = 131072
    const int threads = 256;                // 8 waves per block on wave32
    const int blocks  = (B + threads - 1) / threads;

    ExampleLabelWeights_zero_kernel<<<1, 1, 0, stream>>>(out);
    ExampleLabelWeights_58377195487799_kernel<<<blocks, threads, 0, stream>>>(
        losses, inputs_idx, params, cardinality, out, B);
}